// SNNLayer_30932354466180
// MI455X (gfx1250) — compile-verified
//
#include <hip/hip_runtime.h>
#include <math.h>

#define BATCH   128
#define IN_SIZE 1024
#define OUT_SIZE 512
#define NREAL   1025            // IN_SIZE + 1 (appended ones column)
#define NPAD    2048            // next pow2 for bitonic sort
#define PF      8               // async prefetch depth (ring slots)
#define MAX_SPIKE_TIME 100000.0f

// Issue an async global->LDS copy of 4 bytes for this lane.
// lds_off is the LDS destination, g the 64-bit global source (ASYNCcnt-tracked).
__device__ __forceinline__ void async_b32_to_lds(unsigned lds_off, const float* g) {
    asm volatile("global_load_async_to_lds_b32 %0, %1, off"
                 :: "v"(lds_off), "v"((unsigned long long)(size_t)g)
                 : "memory");
}

__launch_bounds__(512)
__global__ void snn_layer_kernel(const float* __restrict__ layer_in,
                                 const float* __restrict__ weight,
                                 const float* __restrict__ delay,
                                 float* __restrict__ out) {
    // Packed sort keys: [63:32] = IEEE bits of x (all x >= 0 -> monotonic as
    // unsigned), [31:0] = source index. u64 compare == stable ascending argsort.
    __shared__ unsigned long long pairs[NPAD];   // 16 KB
    __shared__ float ring[PF][OUT_SIZE];         // 16 KB
    __shared__ float svals[NREAL + 8];           // unpacked sorted values
    __shared__ int   sidx [NREAL + 8];           // unpacked sorted indices

    const int t = threadIdx.x;   // 0..511, also the output column
    const int b = blockIdx.x;    // batch row

    // ---- Phase 1: x = layer_in * exp(relu(delay)), append 1.0, pad with FLT_MAX
    for (int i = t; i < NPAD; i += 512) {
        float v;
        if (i < IN_SIZE) {
            v = layer_in[(size_t)b * IN_SIZE + i] * __expf(fmaxf(delay[i], 0.0f));
        } else if (i == IN_SIZE) {
            v = 1.0f;            // appended ones column -> weight row 1024
        } else {
            v = __int_as_float(0x7f7fffff);  // FLT_MAX padding, sorts last
        }
        pairs[i] = ((unsigned long long)__float_as_uint(v) << 32) | (unsigned)i;
    }
    __syncthreads();

    // ---- Phase 2: branchless bitonic sort (ascending) of 2048 u64 keys.
    // 512 threads -> 2 disjoint comparators per thread per stage.
    for (int size = 2; size <= NPAD; size <<= 1) {
        for (int stride = size >> 1; stride > 0; stride >>= 1) {
            for (int r = t; r < NPAD / 2; r += 512) {
                int  i   = ((r & ~(stride - 1)) << 1) | (r & (stride - 1));
                int  j   = i + stride;
                bool asc = ((i & size) == 0);
                unsigned long long a  = pairs[i];
                unsigned long long c  = pairs[j];
                unsigned long long lo = (a < c) ? a : c;
                unsigned long long hi = (a < c) ? c : a;
                pairs[i] = asc ? lo : hi;
                pairs[j] = asc ? hi : lo;
            }
            __syncthreads();
        }
    }

    // ---- Unpack the first NREAL sorted pairs for cheap 4-byte broadcast reads.
    for (int i = t; i < NREAL; i += 512) {
        unsigned long long p = pairs[i];
        svals[i] = __uint_as_float((unsigned)(p >> 32));
        sidx[i]  = (int)(unsigned)(p & 0xffffffffu);
    }
    __syncthreads();

    // ---- Phase 3: streaming masked-prefix-sum scan with async-to-LDS prefetch.
    // Lane t loads AND consumes element t of each gathered weight row, so each
    // wave's ASYNCcnt alone orders its slice of the ring: no extra barriers.
    const int      o         = t;
    const unsigned row_bytes = OUT_SIZE * 4u;
    const unsigned lds_base  = (unsigned)(size_t)(&ring[0][o]); // low 32b = LDS offset

    for (int k = 0; k < PF; ++k) {
        int idx = sidx[k];
        async_b32_to_lds(lds_base + (unsigned)k * row_bytes,
                         weight + (size_t)idx * OUT_SIZE + o);
    }

    float wc = 0.0f, wic = 0.0f, best = MAX_SPIKE_TIME;
    float xsk = svals[0];

    int k = 0;
    // Main pipelined loop: always exactly PF rows in flight -> wait asynccnt<=PF-1
    // guarantees (in-order async completion) that row k has landed in LDS.
    for (; k <= NREAL - 1 - PF; ++k) {
        float xsk1 = svals[k + 1];

        asm volatile("s_wait_asynccnt 7" ::: "memory");
        float w = ring[k & (PF - 1)][o];

        wc  += w;
        wic  = fmaf(w, xsk, wic);

        // Refill the slot we just consumed with row k+PF.
        {
            int idxn = sidx[k + PF];
            async_b32_to_lds(lds_base + (unsigned)((k + PF) & (PF - 1)) * row_bytes,
                             weight + (size_t)idxn * OUT_SIZE + o);
        }

        float den  = fmaxf(wc - 1.0f, 1e-10f);
        float cand = wic * __builtin_amdgcn_rcpf(den);
        cand = (wc   < 1.0f) ? MAX_SPIKE_TIME : cand;
        cand = (cand < xsk ) ? MAX_SPIKE_TIME : cand;
        cand = (cand > xsk1) ? MAX_SPIKE_TIME : cand;
        best = fminf(best, cand);

        xsk = xsk1;
    }

    // Tail: drain everything, then finish the last PF rows.
    asm volatile("s_wait_asynccnt 0" ::: "memory");
    for (; k < NREAL; ++k) {
        float xsk1 = (k < NREAL - 1) ? svals[k + 1] : MAX_SPIKE_TIME;
        float w    = ring[k & (PF - 1)][o];

        wc  += w;
        wic  = fmaf(w, xsk, wic);

        float den  = fmaxf(wc - 1.0f, 1e-10f);
        float cand = wic * __builtin_amdgcn_rcpf(den);
        cand = (wc   < 1.0f) ? MAX_SPIKE_TIME : cand;
        cand = (cand < xsk ) ? MAX_SPIKE_TIME : cand;
        cand = (cand > xsk1) ? MAX_SPIKE_TIME : cand;
        best = fminf(best, cand);

        xsk = xsk1;
    }

    out[(size_t)b * OUT_SIZE + o] = best;
}

extern "C" void kernel_launch(void* const* d_in, const int* in_sizes, int n_in,
                              void* d_out, int out_size, void* d_ws, size_t ws_size,
                              hipStream_t stream) {
    const float* layer_in = (const float*)d_in[0]; // (128, 1024)
    const float* weight   = (const float*)d_in[1]; // (1025, 512)
    const float* delay    = (const float*)d_in[2]; // (1024,)
    float*       out      = (float*)d_out;         // (128, 512)

    snn_layer_kernel<<<BATCH, 512, 0, stream>>>(layer_in, weight, delay, out);
}